// SSDRPNHead_3100966388138
// MI455X (gfx1250) — compile-verified
//
#include <hip/hip_runtime.h>

#define CVv 0.1f
#define SVv 0.2f
#define IMGv 300.0f
#define NMS_TH 0.7f
#define KEEP 100
#define EPSv 1e-5f
#define BATCH 16
#define NPRI 16384
#define NCLS 21

typedef __attribute__((ext_vector_type(2))) float v2f;
typedef __attribute__((ext_vector_type(8))) float v8f;

// monotonic float -> uint mapping for atomicMax on floats (handles negatives)
__device__ __forceinline__ unsigned fmap(float f) {
    unsigned u = __float_as_uint(f);
    return (u & 0x80000000u) ? ~u : (u | 0x80000000u);
}
__device__ __forceinline__ float funmap(unsigned u) {
    unsigned b = (u & 0x80000000u) ? (u ^ 0x80000000u) : ~u;
    return __uint_as_float(b);
}

__global__ void init_kernel(unsigned* __restrict__ boxmax_u) {
    if (threadIdx.x < BATCH) boxmax_u[threadIdx.x] = 0u;  // below fmap of any real float
}

// ---------------------------------------------------------------------------
// Kernel 1: decode locations -> pixel corner boxes; per-batch max coordinate.
// grid = B*N/256 blocks of 256 (each block entirely inside one batch).
// ---------------------------------------------------------------------------
__global__ void decode_kernel(const float4* __restrict__ bbox_pred,
                              const float4* __restrict__ priors,
                              float4* __restrict__ boxes,
                              unsigned* __restrict__ boxmax_u) {
    int idx = blockIdx.x * 256 + threadIdx.x;   // over B*N
    int b = idx / NPRI;
    int n = idx - b * NPRI;
    float4 p = priors[n];
    float4 d = bbox_pred[idx];
    float cx = d.x * CVv * p.z + p.x;
    float cy = d.y * CVv * p.w + p.y;
    float w  = expf(d.z * SVv) * p.z;
    float h  = expf(d.w * SVv) * p.w;
    float4 bx;
    bx.x = (cx - w * 0.5f) * IMGv;
    bx.y = (cy - h * 0.5f) * IMGv;
    bx.z = (cx + w * 0.5f) * IMGv;
    bx.w = (cy + h * 0.5f) * IMGv;
    boxes[idx] = bx;

    float m = fmaxf(fmaxf(bx.x, bx.y), fmaxf(bx.z, bx.w));
    __shared__ float red[256];
    red[threadIdx.x] = m;
    __syncthreads();
    for (int s = 128; s > 0; s >>= 1) {
        if (threadIdx.x < s) red[threadIdx.x] = fmaxf(red[threadIdx.x], red[threadIdx.x + s]);
        __syncthreads();
    }
    if (threadIdx.x == 0) atomicMax(&boxmax_u[b], fmap(red[0]));
}

// ---------------------------------------------------------------------------
// Kernel 2: softmax conf/label per anchor. The row-sum of exp(l - max) is
// computed with V_WMMA_F32_16X16X4_F32 (A = exp tile, B = ones): exact f32.
// One wave handles 16 rows; 8 waves / block; grid = B*N/128 blocks.
// ---------------------------------------------------------------------------
__global__ void softmax_wmma_kernel(const float* __restrict__ logits,
                                    float* __restrict__ conf,
                                    int* __restrict__ labels) {
    int wave = threadIdx.x >> 5;
    int lane = threadIdx.x & 31;
    size_t tile = (size_t)blockIdx.x * 8 + wave;        // 16 rows per tile
    size_t row0 = tile * 16;
    size_t myrow = row0 + (size_t)(lane & 15);
    const float* lp = logits + myrow * NCLS;

    // per-row max + first-argmax (both half-waves compute their row's max)
    float m = lp[0];
    int lab = 0;
    #pragma unroll
    for (int c = 1; c < NCLS; c++) {
        float v = lp[c];
        if (v > m) { m = v; lab = c; }
    }
    if (lane < 16) labels[myrow] = lab;

    // A layout for 16x4 f32 WMMA: lanes 0-15 hold K=0,1; lanes 16-31 hold K=2,3
    int khalf = (lane >> 4) << 1;
    v8f acc = {0.f, 0.f, 0.f, 0.f, 0.f, 0.f, 0.f, 0.f};
    v2f ones; ones.x = 1.0f; ones.y = 1.0f;
    #pragma unroll
    for (int j = 0; j < 6; j++) {                        // K-chunks: cols 0..23 (21..23 zero)
        int c0 = j * 4 + khalf;
        int c1 = c0 + 1;
        v2f a;
        a.x = (c0 < NCLS) ? expf(lp[c0] - m) : 0.0f;
        a.y = (c1 < NCLS) ? expf(lp[c1] - m) : 0.0f;
        acc = __builtin_amdgcn_wmma_f32_16x16x4_f32(
            /*neg_a=*/false, a, /*neg_b=*/false, ones,
            /*c_mod=*/(short)0, acc, /*reuse_a=*/false, /*reuse_b=*/false);
    }
    // D layout: lane l<16 holds rows row0+0..7 in acc[0..7]; lane>=16 rows +8..15
    if ((lane & 15) == 0) {
        size_t rb = row0 + (size_t)((lane >> 4) << 3);
        #pragma unroll
        for (int i = 0; i < 8; i++) conf[rb + i] = 1.0f / acc[i];   // conf = exp(0)/sum
    }
}

// ---------------------------------------------------------------------------
// Kernel 3: greedy class-aware NMS, one block (512 threads) per batch.
// Offset boxes staged in 256 KB dynamic LDS; validity = 1 bit/box in regs.
// ---------------------------------------------------------------------------
__global__ void __launch_bounds__(512)
nms_kernel(const float4* __restrict__ boxes, const float* __restrict__ conf,
           const int* __restrict__ labels, const unsigned* __restrict__ boxmax_u,
           int* __restrict__ nms_idx) {
    extern __shared__ float4 ob[];       // NPRI offset boxes: 256 KB
    __shared__ float sval[512];
    __shared__ int   sidx[512];
    __shared__ float4 sbox;
    __shared__ int    sbi;

    int b = blockIdx.x;
    int tid = threadIdx.x;
    const float4* bb = boxes + (size_t)b * NPRI;
    const float*  cf = conf  + (size_t)b * NPRI;
    const int*    lb = labels + (size_t)b * NPRI;
    float off = funmap(boxmax_u[b]) + 1.0f;

    for (int n = tid; n < NPRI; n += 512) {
        float4 v = bb[n];
        float o = (float)lb[n] * off;
        v.x += o; v.y += o; v.z += o; v.w += o;
        ob[n] = v;
    }
    __syncthreads();

    unsigned vb = 0xFFFFFFFFu;           // this thread's 32 boxes: [tid*32, tid*32+32)
    int base = tid * 32;

    for (int k = 0; k < KEEP; k++) {
        // local argmax over my valid boxes (increasing n => strict > keeps first max)
        float bestv = -1.0f;
        int bestn = 0x7FFFFFFF;
        unsigned t = vb;
        while (t) {
            int j = __ffs(t) - 1; t &= t - 1;
            int n = base + j;
            float c = cf[n];
            if (c > bestv) { bestv = c; bestn = n; }
        }
        sval[tid] = bestv; sidx[tid] = bestn;
        __syncthreads();
        for (int s = 256; s > 0; s >>= 1) {
            if (tid < s) {
                float v2 = sval[tid + s]; int i2 = sidx[tid + s];
                if (v2 > sval[tid] || (v2 == sval[tid] && i2 < sidx[tid])) {
                    sval[tid] = v2; sidx[tid] = i2;
                }
            }
            __syncthreads();
        }
        if (tid == 0) {
            int pick = (sval[0] > 0.0f) ? sidx[0] : 0;   // all-invalid => argmax(-inf)=0
            sbi = pick;
            sbox = ob[pick];
            nms_idx[b * KEEP + k] = pick;
        }
        __syncthreads();

        float4 B1 = sbox;
        int pick = sbi;
        float a1 = (B1.z - B1.x) * (B1.w - B1.y);
        t = vb;
        while (t) {
            int j = __ffs(t) - 1; t &= t - 1;
            int n = base + j;
            float4 B2 = ob[n];
            float lx = fmaxf(B1.x, B2.x), ly = fmaxf(B1.y, B2.y);
            float rx = fminf(B1.z, B2.z), ry = fminf(B1.w, B2.w);
            float iw = fmaxf(rx - lx, 0.0f), ih = fmaxf(ry - ly, 0.0f);
            float inter = iw * ih;
            float a2 = (B2.z - B2.x) * (B2.w - B2.y);
            float iou = inter / (a1 + a2 - inter + EPSv);
            if (iou > NMS_TH) vb &= ~(1u << j);
        }
        if (pick >= base && pick < base + 32) vb &= ~(1u << (pick - base));
        // next sval write is separated from neighbors' reads by the barrier above
    }
}

// ---------------------------------------------------------------------------
// Kernel 4: match each kept box to argmax-IoU prior, encode, emit [B,100,8].
// One 256-thread block per (batch, kept-box); priors stay L2-hot.
// ---------------------------------------------------------------------------
__global__ void match_encode_kernel(const float4* __restrict__ boxes,
                                    const float4* __restrict__ priors,
                                    const int* __restrict__ nms_idx,
                                    float* __restrict__ out) {
    int k = blockIdx.x, b = blockIdx.y;
    int tid = threadIdx.x;
    __shared__ float sv[256];
    __shared__ int   si[256];
    __shared__ float4 nbS;
    if (tid == 0) {
        int i = nms_idx[b * KEEP + k];
        nbS = boxes[(size_t)b * NPRI + i];
    }
    __syncthreads();
    float4 pbx = nbS;                                  // pixel corner box
    float4 nb;                                         // normalized (/300)
    nb.x = pbx.x / IMGv; nb.y = pbx.y / IMGv;
    nb.z = pbx.z / IMGv; nb.w = pbx.w / IMGv;
    float a1 = (nb.z - nb.x) * (nb.w - nb.y);

    float bestv = -1.0f;
    int bestn = 0x7FFFFFFF;
    for (int n = tid; n < NPRI; n += 256) {
        __builtin_prefetch(priors + n + 256, 0, 0);    // gfx1250 global_prefetch_b8
        float4 p = priors[n];                          // center form
        float px0 = p.x - p.z * 0.5f, py0 = p.y - p.w * 0.5f;
        float px1 = p.x + p.z * 0.5f, py1 = p.y + p.w * 0.5f;
        float lx = fmaxf(nb.x, px0), ly = fmaxf(nb.y, py0);
        float rx = fminf(nb.z, px1), ry = fminf(nb.w, py1);
        float iw = fmaxf(rx - lx, 0.0f), ih = fmaxf(ry - ly, 0.0f);
        float inter = iw * ih;
        float a2 = (px1 - px0) * (py1 - py0);          // match reference fp exactly
        float iou = inter / (a1 + a2 - inter + EPSv);
        if (iou > bestv) { bestv = iou; bestn = n; }
    }
    sv[tid] = bestv; si[tid] = bestn;
    __syncthreads();
    for (int s = 128; s > 0; s >>= 1) {
        if (tid < s) {
            float v2 = sv[tid + s]; int i2 = si[tid + s];
            if (v2 > sv[tid] || (v2 == sv[tid] && i2 < si[tid])) { sv[tid] = v2; si[tid] = i2; }
        }
        __syncthreads();
    }
    if (tid == 0) {
        float4 pp = priors[si[0]];                     // matched prior (center form)
        float bcx = (nb.x + nb.z) * 0.5f, bcy = (nb.y + nb.w) * 0.5f;
        float bw = nb.z - nb.x, bh = nb.w - nb.y;
        float* o = out + ((size_t)b * KEEP + k) * 8;
        o[0] = pbx.x; o[1] = pbx.y; o[2] = pbx.z; o[3] = pbx.w;
        o[4] = (bcx - pp.x) / pp.z / CVv;
        o[5] = (bcy - pp.y) / pp.w / CVv;
        o[6] = logf(bw / pp.z) / SVv;
        o[7] = logf(bh / pp.w) / SVv;
    }
}

extern "C" void kernel_launch(void* const* d_in, const int* in_sizes, int n_in,
                              void* d_out, int out_size, void* d_ws, size_t ws_size,
                              hipStream_t stream) {
    (void)in_sizes; (void)n_in; (void)out_size; (void)ws_size;
    const float*  cls    = (const float*)d_in[0];          // [B,N,21]
    const float4* bbox   = (const float4*)d_in[1];         // [B,N,4]
    const float4* priors = (const float4*)d_in[2];         // [N,4] center form

    char* ws = (char*)d_ws;
    float4* boxes = (float4*)ws;                           // B*N float4   (4 MiB)
    size_t o = (size_t)BATCH * NPRI * sizeof(float4);
    float* conf = (float*)(ws + o);  o += (size_t)BATCH * NPRI * sizeof(float);
    int* labels = (int*)(ws + o);    o += (size_t)BATCH * NPRI * sizeof(int);
    unsigned* boxmax_u = (unsigned*)(ws + o); o += 256;
    int* nms_idx = (int*)(ws + o);
    float* out = (float*)d_out;

    init_kernel<<<1, 32, 0, stream>>>(boxmax_u);
    decode_kernel<<<(BATCH * NPRI) / 256, 256, 0, stream>>>(bbox, priors, boxes, boxmax_u);
    softmax_wmma_kernel<<<(BATCH * NPRI) / 128, 256, 0, stream>>>(cls, conf, labels);
    nms_kernel<<<BATCH, 512, NPRI * sizeof(float4), stream>>>(boxes, conf, labels, boxmax_u, nms_idx);
    match_encode_kernel<<<dim3(KEEP, BATCH), 256, 0, stream>>>(boxes, priors, nms_idx, out);
}